// AmbiguityHead_45938970198650
// MI455X (gfx1250) — compile-verified
//
#include <hip/hip_runtime.h>
#include <stdint.h>

// ---------------------------------------------------------------------------
// AmbiguityHead for MI455X (gfx1250, wave32)
//
// Two passes:
//   1) argmax13: labels[N,13] -> uint8 label[N] in d_ws   (kills 52B gathers)
//   2) ambiguity: per point, gather 15 neighbor (xyz, label-byte), softmax
// Regular slabs (neighbor_idx / p / labels tiles) are DMA'd into LDS with the
// CDNA5 Tensor Data Mover (TENSOR_LOAD_TO_LDS + s_wait_tensorcnt), with a
// cooperative-load fallback if the builtin is unavailable in this toolchain.
// ---------------------------------------------------------------------------

typedef __attribute__((ext_vector_type(4))) unsigned int u32x4;
typedef __attribute__((ext_vector_type(8))) int          i32x8;
typedef __attribute__((ext_vector_type(4))) int          i32x4;

static constexpr int   CLS    = 13;   // num classes
static constexpr int   KNN    = 15;   // neighbors per point
static constexpr float CCBETA = 2.0f;
static constexpr float NU     = 1.0f;
static constexpr int   BLK    = 256;  // 8 waves of 32 on CDNA5

#if defined(__HIP_DEVICE_COMPILE__) && __has_builtin(__builtin_amdgcn_tensor_load_to_lds)
#define HAVE_TDM 1
#else
#define HAVE_TDM 0
#endif

#if HAVE_TDM
// Low 32 bits of a flat shared-aperture address are the LDS byte offset
// (CDNA5 ISA aperture table).
__device__ __forceinline__ uint32_t lds_offset(const void* smem_ptr) {
  return (uint32_t)(uintptr_t)smem_ptr;
}

// 1-D TDM copy: ndwords DWORDs from global `gptr` into LDS byte offset.
// D# layout per CDNA5 ISA.
__device__ __forceinline__ void tdm_load_1d(uint32_t lds_byte_off,
                                            const void* gptr,
                                            uint32_t ndwords) {
  const uint64_t ga = (uint64_t)(uintptr_t)gptr;

  u32x4 g0;
  g0[0] = 1u;                                   // count=1, user mode, no gather
  g0[1] = lds_byte_off;                         // lds_addr
  g0[2] = (uint32_t)ga;                         // global_addr[31:0]
  g0[3] = (uint32_t)((ga >> 32) & 0x01FFFFFFu)  // global_addr[56:32]
        | (2u << 30);                           // type = 2 ("image")

  i32x8 g1;
  g1[0] = (int)(2u << 16);                      // wg_mask=0, data_size=2 (4B)
  g1[1] = (int)((ndwords & 0xFFFFu) << 16);     // tensor_dim0[15:0]
  g1[2] = (int)(((ndwords >> 16) & 0xFFFFu)     // tensor_dim0[31:16]
        | (1u << 16));                          // tensor_dim1 = 1
  g1[3] = (int)((ndwords & 0xFFFFu) << 16);     // tile_dim0 = ndwords (<=65535)
  g1[4] = (int)(1u | (1u << 16));               // tile_dim1 = 1, tile_dim2 = 1
  g1[5] = (int)ndwords;                         // tensor_dim0_stride lo32
  g1[6] = 0;                                    // stride0 hi, stride1 lo
  g1[7] = 0;                                    // stride1 hi

  i32x4 g2;
  g2[0] = 1;                                    // tensor_dim2 = 1
  g2[1] = 1;                                    // tensor_dim3 = 1
  g2[2] = 0;                                    // dim2 stride lo
  g2[3] = (int)(1u << 16);                      // tile_dim3 = 1

  i32x4 g3;
  g3[0] = 0;                                    // dim3 stride lo
  g3[1] = (int)(1u << 16);                      // tensor_dim4 = 1
  g3[2] = (int)(1u << 16);                      // tile_dim4 = 1
  g3[3] = 0;

#if __clang_major__ >= 23
  i32x8 gz = {0, 0, 0, 0, 0, 0, 0, 0};
  __builtin_amdgcn_tensor_load_to_lds(g0, g1, g2, g3, gz, 0);
#else
  __builtin_amdgcn_tensor_load_to_lds(g0, g1, g2, g3, 0);
#endif
}
#endif  // HAVE_TDM

// ---------------------------------------------------------------------------
// Pass 1: per-point argmax over 13 classes -> uint8 label
// ---------------------------------------------------------------------------
__global__ void __launch_bounds__(BLK)
argmax13_kernel(const float* __restrict__ labels,
                unsigned char* __restrict__ lab,
                int n) {
  __shared__ float s_lab[BLK * CLS];  // 13312 B

  const int tid  = threadIdx.x;
  const int base = blockIdx.x * BLK;
  const int rem  = n - base;
  const int npts = rem < BLK ? rem : BLK;
  const uint32_t nd = (uint32_t)(npts * CLS);

#if HAVE_TDM
  if (tid == 0) {
    tdm_load_1d(lds_offset(s_lab), labels + (size_t)base * CLS, nd);
    __builtin_amdgcn_s_wait_tensorcnt(0);
  }
#else
  {
    const float* src = labels + (size_t)base * CLS;
    for (uint32_t k = tid; k < nd; k += BLK) s_lab[k] = src[k];
  }
#endif
  __syncthreads();

  const int i = base + tid;
  if (i < n) {
    float best = s_lab[tid * CLS];
    int bi = 0;
#pragma unroll
    for (int c = 1; c < CLS; ++c) {
      const float v = s_lab[tid * CLS + c];
      if (v > best) { best = v; bi = c; }  // strict '>' => first-max (jax argmax)
    }
    lab[i] = (unsigned char)bi;
  }
}

// ---------------------------------------------------------------------------
// Pass 2: distance-weighted soft fraction of negative neighbors
// ---------------------------------------------------------------------------
__global__ void __launch_bounds__(BLK)
ambiguity_kernel(const float* __restrict__ p,
                 const int* __restrict__ nidx,
                 const unsigned char* __restrict__ lab,
                 float* __restrict__ out,
                 int n) {
  __shared__ int   s_idx[BLK * KNN];  // 15360 B
  __shared__ float s_p[BLK * 3];      //  3072 B

  const int tid  = threadIdx.x;
  const int base = blockIdx.x * BLK;
  const int rem  = n - base;
  const int npts = rem < BLK ? rem : BLK;

#if HAVE_TDM
  if (tid == 0) {
    tdm_load_1d(lds_offset(s_idx), nidx + (size_t)base * KNN,
                (uint32_t)(npts * KNN));
    tdm_load_1d(lds_offset(s_p), p + (size_t)base * 3,
                (uint32_t)(npts * 3));
    __builtin_amdgcn_s_wait_tensorcnt(0);
  }
#else
  {
    const int*   si = nidx + (size_t)base * KNN;
    const float* sp = p + (size_t)base * 3;
    const uint32_t ni = (uint32_t)(npts * KNN);
    const uint32_t np3 = (uint32_t)(npts * 3);
    for (uint32_t k = tid; k < ni; k += BLK)  s_idx[k] = si[k];
    for (uint32_t k = tid; k < np3; k += BLK) s_p[k]   = sp[k];
  }
#endif
  __syncthreads();

  const int i = base + tid;
  if (i < n) {
    const float px = s_p[tid * 3 + 0];
    const float py = s_p[tid * 3 + 1];
    const float pz = s_p[tid * 3 + 2];
    const int   cl = lab[i];  // coalesced byte load

    float d2[KNN];
    unsigned negbits = 0;
#pragma unroll
    for (int j = 0; j < KNN; ++j) {
      const int nb = s_idx[tid * KNN + j];
      const size_t b3 = (size_t)nb * 3;
      const float qx = p[b3 + 0];   // random gather, L2-resident (6 MB)
      const float qy = p[b3 + 1];
      const float qz = p[b3 + 2];
      const int   nl = lab[nb];     // random 1-byte gather, L2-resident (0.5 MB)
      const float dx = px - qx, dy = py - qy, dz = pz - qz;
      d2[j] = dx * dx + dy * dy + dz * dz;
      negbits |= (nl != cl) ? (1u << j) : 0u;
    }

    // softmax(-beta*d2) via stable form exp(-beta*(d2 - min(d2)))
    float dmin = d2[0];
#pragma unroll
    for (int j = 1; j < KNN; ++j) dmin = fminf(dmin, d2[j]);

    float denom = 0.0f, num = 0.0f;
#pragma unroll
    for (int j = 0; j < KNN; ++j) {
      const float e = __expf(-CCBETA * (d2[j] - dmin));  // v_exp_f32 path
      denom += e;
      num   += ((negbits >> j) & 1u) ? e : 0.0f;
    }
    out[i] = NU * (num / denom);
  }
}

// ---------------------------------------------------------------------------
// Launch
// ---------------------------------------------------------------------------
extern "C" void kernel_launch(void* const* d_in, const int* in_sizes, int n_in,
                              void* d_out, int out_size, void* d_ws, size_t ws_size,
                              hipStream_t stream) {
  const float* p      = (const float*)d_in[0];  // [N,3]
  const float* labels = (const float*)d_in[1];  // [N,13]
  const int*   nidx   = (const int*)d_in[2];    // [N,15]
  float*       out    = (float*)d_out;          // [N]

  const int n = in_sizes[0] / 3;
  unsigned char* lab = (unsigned char*)d_ws;    // n bytes of scratch

  const int blocks = (n + BLK - 1) / BLK;
  argmax13_kernel<<<blocks, BLK, 0, stream>>>(labels, lab, n);
  ambiguity_kernel<<<blocks, BLK, 0, stream>>>(p, nidx, lab, out, n);
}